// MultiHeadAttention_70549132804637
// MI455X (gfx1250) — compile-verified
//
#include <hip/hip_runtime.h>

// Problem constants (from reference): B=4, S=2048, E=1024, H=16, D=64
#define BB 4
#define SS 2048
#define EE 1024
#define HH 16
#define DD 64
#define E3 3072
#define SCALE_F 0.125f  // 1/sqrt(64)

typedef __attribute__((ext_vector_type(2))) float v2f;
typedef __attribute__((ext_vector_type(8))) float v8f;

__device__ __forceinline__ v8f vzero8() {
  v8f z;
#pragma unroll
  for (int i = 0; i < 8; ++i) z[i] = 0.0f;
  return z;
}

// D = A(16x4) * B(4x16) + C  -- CDNA5 fp32 WMMA
__device__ __forceinline__ v8f wmma4(v2f a, v2f b, v8f c) {
  return __builtin_amdgcn_wmma_f32_16x16x4_f32(
      /*neg_a=*/false, a, /*neg_b=*/false, b,
      /*c_mod=*/(short)0, c, /*reuse_a=*/false, /*reuse_b=*/false);
}

// -------------------------------------------------------------------------
// C[M,N] = A[M,K] * W[N,K]^T   (row-major; M%128==0, N%64==0, K%32==0)
// block: 256 threads = 8 waves; block tile 128(M) x 64(N); wave tile 32x32.
// -------------------------------------------------------------------------
__global__ __launch_bounds__(256) void gemm_nt_f32(
    const float* __restrict__ A, const float* __restrict__ W,
    float* __restrict__ C, int M, int N, int K) {
  __shared__ float As[128][33];
  __shared__ float Ws[64][33];

  const int t    = threadIdx.x;
  const int lane = t & 31;
  const int wv   = t >> 5;   // 0..7
  const int wm   = wv & 3;   // M quadrant (32 rows each)
  const int wn   = wv >> 2;  // N half (32 cols each)

  const int m0 = blockIdx.y * 128;
  const int n0 = blockIdx.x * 64;

  const int r    = lane & 15;
  const int hi   = lane >> 4;  // 0 or 1
  const int koff = hi * 2;

  v8f acc[2][2];
#pragma unroll
  for (int i = 0; i < 2; ++i)
#pragma unroll
    for (int j = 0; j < 2; ++j) acc[i][j] = vzero8();

  for (int k0 = 0; k0 < K; k0 += 32) {
    // Stage A tile: 128x32 floats, 16 per thread (4x float4)
    {
      const int row = t >> 1;
      const int kb  = (t & 1) * 16;
      const float4* src =
          (const float4*)(A + (size_t)(m0 + row) * K + k0 + kb);
#pragma unroll
      for (int q = 0; q < 4; ++q) {
        float4 v = src[q];
        As[row][kb + 4 * q + 0] = v.x;
        As[row][kb + 4 * q + 1] = v.y;
        As[row][kb + 4 * q + 2] = v.z;
        As[row][kb + 4 * q + 3] = v.w;
      }
    }
    // Stage W tile: 64x32 floats, 8 per thread (2x float4)
    {
      const int nrow = t >> 2;
      const int kb   = (t & 3) * 8;
      const float4* src =
          (const float4*)(W + (size_t)(n0 + nrow) * K + k0 + kb);
#pragma unroll
      for (int q = 0; q < 2; ++q) {
        float4 v = src[q];
        Ws[nrow][kb + 4 * q + 0] = v.x;
        Ws[nrow][kb + 4 * q + 1] = v.y;
        Ws[nrow][kb + 4 * q + 2] = v.z;
        Ws[nrow][kb + 4 * q + 3] = v.w;
      }
    }
    __syncthreads();

#pragma unroll
    for (int kk = 0; kk < 32; kk += 4) {
      v2f af[2], bf[2];
#pragma unroll
      for (int i = 0; i < 2; ++i) {
        const int arow = wm * 32 + i * 16 + r;
        af[i].x = As[arow][kk + koff];
        af[i].y = As[arow][kk + koff + 1];
      }
#pragma unroll
      for (int j = 0; j < 2; ++j) {
        const int ncol = wn * 32 + j * 16 + r;  // B[k][n] = W[n][k]
        bf[j].x = Ws[ncol][kk + koff];
        bf[j].y = Ws[ncol][kk + koff + 1];
      }
#pragma unroll
      for (int i = 0; i < 2; ++i)
#pragma unroll
        for (int j = 0; j < 2; ++j) acc[i][j] = wmma4(af[i], bf[j], acc[i][j]);
    }
    __syncthreads();
  }

#pragma unroll
  for (int i = 0; i < 2; ++i)
#pragma unroll
    for (int j = 0; j < 2; ++j) {
      const int col = n0 + wn * 32 + j * 16 + r;
#pragma unroll
      for (int r8 = 0; r8 < 8; ++r8) {
        const int row = m0 + wm * 32 + i * 16 + r8 + hi * 8;
        C[(size_t)row * N + col] = acc[i][j][r8];
      }
    }
}

// -------------------------------------------------------------------------
// Per (b,h): KV[d1][d2] = sum_s K[s][d1] * V[s][d2]   (64x64 output)
// K[s][d] = qkv[(b*S+s)*3E + h*3D + D + d], V[s][d] = ... + 2D + d
// block: 256 threads; wave w handles tile (row w/2, cols 2*(w&1)..)
// -------------------------------------------------------------------------
__global__ __launch_bounds__(256) void ktv_f32(const float* __restrict__ qkv,
                                               float* __restrict__ kvb) {
  __shared__ float Ks[64][65];
  __shared__ float Vs[64][65];

  const int t    = threadIdx.x;
  const int lane = t & 31;
  const int wv   = t >> 5;
  const int tr   = wv >> 1;         // output tile row 0..3
  const int tc0  = (wv & 1) * 2;    // output tile cols tc0, tc0+1
  const int bh   = blockIdx.x;
  const int b    = bh >> 4;
  const int h    = bh & 15;

  const int r    = lane & 15;
  const int hi   = lane >> 4;
  const int koff = hi * 2;

  v8f acc0 = vzero8(), acc1 = vzero8();

  const size_t base = (size_t)b * SS * E3 + (size_t)h * (3 * DD);

  for (int s0 = 0; s0 < SS; s0 += 64) {
    const int srow = t >> 2;        // 0..63
    const int db   = (t & 3) * 16;
    const float* kp = qkv + base + (size_t)(s0 + srow) * E3 + DD + db;
    const float* vp = qkv + base + (size_t)(s0 + srow) * E3 + 2 * DD + db;
#pragma unroll
    for (int q = 0; q < 4; ++q) {
      float4 vk = ((const float4*)kp)[q];
      float4 vv = ((const float4*)vp)[q];
      Ks[srow][db + 4 * q + 0] = vk.x; Ks[srow][db + 4 * q + 1] = vk.y;
      Ks[srow][db + 4 * q + 2] = vk.z; Ks[srow][db + 4 * q + 3] = vk.w;
      Vs[srow][db + 4 * q + 0] = vv.x; Vs[srow][db + 4 * q + 1] = vv.y;
      Vs[srow][db + 4 * q + 2] = vv.z; Vs[srow][db + 4 * q + 3] = vv.w;
    }
    __syncthreads();

#pragma unroll
    for (int kk = 0; kk < 64; kk += 4) {
      v2f af, b0, b1;
      // A = K^T: A[m][k] = Ks[k][m] -> transposed read from LDS (free)
      af.x = Ks[kk + koff][tr * 16 + r];
      af.y = Ks[kk + koff + 1][tr * 16 + r];
      b0.x = Vs[kk + koff][tc0 * 16 + r];
      b0.y = Vs[kk + koff + 1][tc0 * 16 + r];
      b1.x = Vs[kk + koff][(tc0 + 1) * 16 + r];
      b1.y = Vs[kk + koff + 1][(tc0 + 1) * 16 + r];
      acc0 = wmma4(af, b0, acc0);
      acc1 = wmma4(af, b1, acc1);
    }
    __syncthreads();
  }

  float* out = kvb + (size_t)bh * DD * DD;
#pragma unroll
  for (int r8 = 0; r8 < 8; ++r8) {
    const int row = tr * 16 + r8 + hi * 8;
    out[row * DD + tc0 * 16 + r]       = acc0[r8];
    out[row * DD + (tc0 + 1) * 16 + r] = acc1[r8];
  }
}

// -------------------------------------------------------------------------
// Per (b,h, 128-row slab): values[b,s,h*D+d] = SCALE * sum_d1 Q[s][d1]*KV[d1][d]
// Q[s][d1] = qkv[(b*S+s)*3E + h*3D + d1]
// block: 256 threads; wave w computes 16 rows x full 64 cols (4 WMMA tiles)
// -------------------------------------------------------------------------
__global__ __launch_bounds__(256) void q_kv_f32(const float* __restrict__ qkv,
                                                const float* __restrict__ kvb,
                                                float* __restrict__ values) {
  __shared__ float Qs[128][65];
  __shared__ float KVs[64][65];

  const int t    = threadIdx.x;
  const int lane = t & 31;
  const int wv   = t >> 5;  // row tile 0..7
  const int bh   = blockIdx.y;
  const int b    = bh >> 4;
  const int h    = bh & 15;
  const int s0   = blockIdx.x * 128;

  const int r    = lane & 15;
  const int hi   = lane >> 4;
  const int koff = hi * 2;

  // Stage KV (64x64): 16 floats per thread
  {
    const int row = t >> 2;
    const int db  = (t & 3) * 16;
    const float4* src = (const float4*)(kvb + (size_t)bh * 4096 + row * DD + db);
#pragma unroll
    for (int q = 0; q < 4; ++q) {
      float4 v = src[q];
      KVs[row][db + 4 * q + 0] = v.x; KVs[row][db + 4 * q + 1] = v.y;
      KVs[row][db + 4 * q + 2] = v.z; KVs[row][db + 4 * q + 3] = v.w;
    }
  }
  // Stage Q slab (128x64): 32 floats per thread
  {
    const int row = t >> 1;
    const int db  = (t & 1) * 32;
    const float4* src = (const float4*)(
        qkv + (size_t)(b * SS + s0 + row) * E3 + h * 3 * DD + db);
#pragma unroll
    for (int q = 0; q < 8; ++q) {
      float4 v = src[q];
      Qs[row][db + 4 * q + 0] = v.x; Qs[row][db + 4 * q + 1] = v.y;
      Qs[row][db + 4 * q + 2] = v.z; Qs[row][db + 4 * q + 3] = v.w;
    }
  }
  __syncthreads();

  v8f acc[4];
#pragma unroll
  for (int j = 0; j < 4; ++j) acc[j] = vzero8();

#pragma unroll
  for (int kk = 0; kk < 64; kk += 4) {
    v2f af;
    af.x = Qs[wv * 16 + r][kk + koff];
    af.y = Qs[wv * 16 + r][kk + koff + 1];
#pragma unroll
    for (int j = 0; j < 4; ++j) {
      v2f bf;
      bf.x = KVs[kk + koff][j * 16 + r];
      bf.y = KVs[kk + koff + 1][j * 16 + r];
      acc[j] = wmma4(af, bf, acc[j]);
    }
  }

#pragma unroll
  for (int j = 0; j < 4; ++j) {
    const int col = j * 16 + r;
#pragma unroll
    for (int r8 = 0; r8 < 8; ++r8) {
      const int row = wv * 16 + r8 + hi * 8;
      values[(size_t)(b * SS + s0 + row) * EE + h * DD + col] =
          SCALE_F * acc[j][r8];
    }
  }
}

// -------------------------------------------------------------------------
extern "C" void kernel_launch(void* const* d_in, const int* in_sizes, int n_in,
                              void* d_out, int out_size, void* d_ws,
                              size_t ws_size, hipStream_t stream) {
  (void)in_sizes; (void)n_in; (void)out_size; (void)ws_size;
  const float* x     = (const float*)d_in[0];
  const float* w_qkv = (const float*)d_in[1];
  const float* w_o   = (const float*)d_in[2];
  float* out = (float*)d_out;

  char* ws = (char*)d_ws;
  const size_t qkv_bytes = (size_t)BB * SS * E3 * sizeof(float);      // 100.7 MB
  const size_t kv_bytes  = (size_t)BB * HH * DD * DD * sizeof(float); // 1 MB
  float* qkv    = (float*)ws;
  float* kvb    = (float*)(ws + qkv_bytes);
  float* values = (float*)(ws + qkv_bytes + kv_bytes);                // 33.5 MB

  const int M = BB * SS;  // 8192 token rows

  // 1) qkv = x @ w_qkv^T            (8192 x 3072 x 1024)
  gemm_nt_f32<<<dim3(E3 / 64, M / 128), 256, 0, stream>>>(x, w_qkv, qkv, M, E3, EE);
  // 2) KV[b,h] = K^T V              (64 heads, 64x64x2048 reductions)
  ktv_f32<<<dim3(BB * HH), 256, 0, stream>>>(qkv, kvb);
  // 3) values = scale * Q @ KV      (64 heads, 2048x64x64)
  q_kv_f32<<<dim3(SS / 128, BB * HH), 256, 0, stream>>>(qkv, kvb, values);
  // 4) out = values @ w_o^T         (8192 x 1024 x 1024)
  gemm_nt_f32<<<dim3(EE / 64, M / 128), 256, 0, stream>>>(values, w_o, out, M, EE, EE);
}